// GroupPointConv_91293824844262
// MI455X (gfx1250) — compile-verified
//
#include <hip/hip_runtime.h>

// ---------------------------------------------------------------------------
// Problem constants (match reference)
// ---------------------------------------------------------------------------
constexpr int Bn   = 8;
constexpr int Nn   = 2048;
constexpr int Kn   = 16;          // group size == neighbor count
constexpr int CIN  = 64;
constexpr int CMID = 16;
constexpr int COUT = 64;
constexpr int FH1  = 256;         // final hidden
constexpr int KDIM = CMID * CIN;  // 1024

// WMMA fragment types (f32 16x16x4: A/B = 2 f32 per lane, C/D = 8 f32 per lane)
typedef float v2f __attribute__((ext_vector_type(2)));
typedef float v8f __attribute__((ext_vector_type(8)));

__device__ __forceinline__ v8f wmma_f32(v2f a, v2f b, v8f c) {
  // D = A(16x4) * B(4x16) + C  in fp32 on the matrix pipe
  return __builtin_amdgcn_wmma_f32_16x16x4_f32(
      /*neg_a=*/false, a, /*neg_b=*/false, b,
      /*c_mod=*/(short)0, c, /*reuse_a=*/false, /*reuse_b=*/false);
}

// ---------------------------------------------------------------------------
// LDS layout (floats). Total 14048 floats = 56,192 B (static, < 64KB).
// ---------------------------------------------------------------------------
constexpr int FSTR    = 18;                   // F_T row pad: 18*m mod 64 distinct -> no bank conflicts
constexpr int OFF_F   = 0;                    // F_T[c][k]  : 64*18  = 1152
constexpr int OFF_M   = 1152;                 // M[m][n][k] : 256*18 = 4608
constexpr int OFF_PTS = 5760;                 // pts[16][3] : 48
constexpr int OFF_MSK = 5808;                 // mask[16]   : 16
constexpr int ESTR    = 514;                  // e row pad: mod 64 == 2 -> skewed banks
constexpr int OFF_E   = 5824;                 // e_half[16][514] = 8224
constexpr int SMEM_F  = 14048;
constexpr int HSTR    = 258;                  // h row pad (reuses M region: 16*258=4128 <= 4608)
constexpr int OFF_H   = OFF_M;

// ---------------------------------------------------------------------------
// Pack f1 [1024x256] and f2 [256x64] into per-lane B-fragment order so the
// GEMM inner loop does a single coalesced global_load_b64 per fragment.
// pair index = ((k/4)*numJTiles + jt)*32 + lane ;
// lane<16 -> rows k+{0,1}, col jt*16+lane ; lane>=16 -> rows k+{2,3}.
// ---------------------------------------------------------------------------
__global__ __launch_bounds__(256) void pack_weights(
    const float* __restrict__ f1, const float* __restrict__ f2,
    float* __restrict__ f1p, float* __restrict__ f2p) {
  int tid = blockIdx.x * 256 + threadIdx.x;
  constexpr int NP1 = (KDIM / 4) * (FH1 / 16) * 32;   // 131072 pairs
  constexpr int NP2 = (FH1 / 4) * (COUT / 16) * 32;   // 8192 pairs
  if (tid < NP1) {
    int lane = tid & 31, jt = (tid >> 5) & 15, kk4 = tid >> 9;
    int c = lane & 15, dk = (lane >> 4) << 1;
    int r0 = kk4 * 4 + dk;
    f1p[tid * 2 + 0] = f1[(r0 + 0) * FH1 + jt * 16 + c];
    f1p[tid * 2 + 1] = f1[(r0 + 1) * FH1 + jt * 16 + c];
  } else if (tid < NP1 + NP2) {
    int t2 = tid - NP1;
    int lane = t2 & 31, ot = (t2 >> 5) & 3, kk4 = t2 >> 7;
    int c = lane & 15, dk = (lane >> 4) << 1;
    int r0 = kk4 * 4 + dk;
    f2p[t2 * 2 + 0] = f2[(r0 + 0) * COUT + ot * 16 + c];
    f2p[t2 * 2 + 1] = f2[(r0 + 1) * COUT + ot * 16 + c];
  }
}

// ---------------------------------------------------------------------------
// One block per (group g, batch b). 256 threads = 8 waves (wave32).
// ---------------------------------------------------------------------------
__global__ __launch_bounds__(256) void gpc_main(
    const float* __restrict__ points, const float* __restrict__ feats,
    const float* __restrict__ pmask,
    const float* __restrict__ w1, const float* __restrict__ b1,
    const float* __restrict__ w2, const float* __restrict__ b2,
    const float* __restrict__ w3, const float* __restrict__ b3,
    const float* __restrict__ fb1, const float* __restrict__ fb2,
    const float* __restrict__ f1p, const float* __restrict__ f2p,
    float* __restrict__ out) {
  __shared__ float smem[SMEM_F];
  float* Fl = smem + OFF_F;    // F_T[c][k]  (feats * mask, transposed)
  float* Ml = smem + OFF_M;    // M[m][n][k]
  float* Pt = smem + OFF_PTS;
  float* Mk = smem + OFF_MSK;
  float* El = smem + OFF_E;    // e half-buffer [n][512]
  float* Hl = smem + OFF_H;    // h[n][256] (aliases Ml after last read)

  const int g = blockIdx.x;            // group 0..127
  const int b = blockIdx.y;            // batch 0..7
  const int t = threadIdx.x;
  const int wave = t >> 5;
  const int lane = t & 31;
  const int lanelo = lane & 15;
  const int dk = (lane >> 4) << 1;     // K sub-offset for A/B fragments: 0 or 2
  const int mbase = (lane >> 4) << 3;  // C/D row base: 0 or 8
  const int base = b * Nn + g * Kn;    // first point of this group

  // ---- Phase 0: stage points/mask/masked-feats into LDS -------------------
  if (t < 48) Pt[t] = points[(size_t)base * 3 + t];
  if (t >= 48 && t < 64) Mk[t - 48] = pmask[base + (t - 48)];
  {
    int k = t >> 4, c0 = (t & 15) * 4;
    float mk = pmask[base + k];
    const float4 v = *(const float4*)(feats + (size_t)(base + k) * CIN + c0);
    Fl[(c0 + 0) * FSTR + k] = v.x * mk;
    Fl[(c0 + 1) * FSTR + k] = v.y * mk;
    Fl[(c0 + 2) * FSTR + k] = v.z * mk;
    Fl[(c0 + 3) * FSTR + k] = v.w * mk;
  }
  __syncthreads();

  // ---- Phase 1: pointwise MLP 3->32->32->16 (ReLU), masked ----------------
  // One (n,k) pair per thread; weight loads are wave-uniform -> scalar path.
  {
    int n = t >> 4, k = t & 15;
    float rx = Pt[n * 3 + 0] - Pt[k * 3 + 0];
    float ry = Pt[n * 3 + 1] - Pt[k * 3 + 1];
    float rz = Pt[n * 3 + 2] - Pt[k * 3 + 2];
    float h1[32];
#pragma unroll
    for (int j = 0; j < 32; ++j)
      h1[j] = fmaxf(0.f, rx * w1[j] + ry * w1[32 + j] + rz * w1[64 + j] + b1[j]);
    float h2[32];
#pragma unroll
    for (int j = 0; j < 32; ++j) {
      float s = b2[j];
#pragma unroll
      for (int i = 0; i < 32; ++i) s += h1[i] * w2[i * 32 + j];
      h2[j] = fmaxf(0.f, s);
    }
    float mk = Mk[k];
#pragma unroll
    for (int j = 0; j < 16; ++j) {
      float s = b3[j];
#pragma unroll
      for (int i = 0; i < 32; ++i) s += h2[i] * w3[i * 16 + j];
      Ml[(j * 16 + n) * FSTR + k] = fmaxf(0.f, s) * mk;   // M[m=j][n][k]
    }
  }
  __syncthreads();

  // ---- Phases 2+3 interleaved over two K-halves of e ----------------------
  // Layer-1 accumulators persist across halves (2 N-tiles per wave).
  const int jt0 = wave, jt1 = wave + 8;
  const float fb1v0 = fb1[jt0 * 16 + lanelo];
  const float fb1v1 = fb1[jt1 * 16 + lanelo];
  v8f acc0 = {};
  v8f acc1 = {};

  for (int half = 0; half < 2; ++half) {
    // -- einsum: e[n][m*64+c] = sum_k M[n][k][m] * F[k][c]; wave owns m.
    const int m = half * 8 + wave;          // absolute m handled by this wave
    const int ml = wave;                    // m local to this half
    for (int ct = 0; ct < 4; ++ct) {        // 4 column tiles of CIN=64
      v8f e = {};
#pragma unroll
      for (int kk = 0; kk < 16; kk += 4) {
        v2f a  = *(const v2f*)&Ml[(m * 16 + lanelo) * FSTR + kk + dk];
        v2f bb = *(const v2f*)&Fl[(ct * 16 + lanelo) * FSTR + kk + dk];
        e = wmma_f32(a, bb, e);
      }
#pragma unroll
      for (int r = 0; r < 8; ++r)
        El[(mbase + r) * ESTR + ml * 64 + ct * 16 + lanelo] = e[r];
    }
    __syncthreads();

    // -- layer 1 partial GEMM: [16 x 512(half)] @ f1p -> acc
#pragma unroll 4
    for (int kk4 = 0; kk4 < 128; ++kk4) {
      int kkh = kk4 * 4;
      int ka = half * 128 + kk4;            // absolute K/4 index into f1p
      v2f a  = *(const v2f*)&El[lanelo * ESTR + kkh + dk];
      v2f b0 = *(const v2f*)&f1p[(size_t)((ka * 16 + jt0) * 32 + lane) * 2];
      v2f b1v = *(const v2f*)&f1p[(size_t)((ka * 16 + jt1) * 32 + lane) * 2];
      acc0 = wmma_f32(a, b0, acc0);
      acc1 = wmma_f32(a, b1v, acc1);
    }
    __syncthreads();                        // El (and, after half 1, Ml) free
  }

  // ---- bias + ReLU, stash h into LDS (reuses M region) --------------------
#pragma unroll
  for (int r = 0; r < 8; ++r) {
    int n = mbase + r;
    Hl[n * HSTR + jt0 * 16 + lanelo] = fmaxf(0.f, acc0[r] + fb1v0);
    Hl[n * HSTR + jt1 * 16 + lanelo] = fmaxf(0.f, acc1[r] + fb1v1);
  }
  __syncthreads();

  // ---- layer 2: [16 x 256] @ f2 -> [16 x 64], waves 0..3 ------------------
  if (wave < 4) {
    const int ot = wave;
    const float fb2v = fb2[ot * 16 + lanelo];
    v8f o = {};
#pragma unroll 4
    for (int kk4 = 0; kk4 < 64; ++kk4) {
      v2f a  = *(const v2f*)&Hl[lanelo * HSTR + kk4 * 4 + dk];
      v2f bb = *(const v2f*)&f2p[(size_t)((kk4 * 4 + ot) * 32 + lane) * 2];
      o = wmma_f32(a, bb, o);
    }
#pragma unroll
    for (int r = 0; r < 8; ++r)
      out[(size_t)(base + mbase + r) * COUT + ot * 16 + lanelo] = o[r] + fb2v;
  }
}

// ---------------------------------------------------------------------------
// Harness entry point
// ---------------------------------------------------------------------------
extern "C" void kernel_launch(void* const* d_in, const int* in_sizes, int n_in,
                              void* d_out, int out_size, void* d_ws, size_t ws_size,
                              hipStream_t stream) {
  (void)in_sizes; (void)n_in; (void)out_size; (void)ws_size;
  // setup_inputs() order:
  // 0 groups(i32) 1 points 2 feats 3 point_mask 4 w1 5 b1 6 w2 7 b2 8 w3 9 b3
  // 10 f1 11 fb1 12 f2 13 fb2
  const float* points = (const float*)d_in[1];
  const float* feats  = (const float*)d_in[2];
  const float* pmask  = (const float*)d_in[3];
  const float* w1 = (const float*)d_in[4];
  const float* b1 = (const float*)d_in[5];
  const float* w2 = (const float*)d_in[6];
  const float* b2 = (const float*)d_in[7];
  const float* w3 = (const float*)d_in[8];
  const float* b3 = (const float*)d_in[9];
  const float* f1  = (const float*)d_in[10];
  const float* fb1 = (const float*)d_in[11];
  const float* f2  = (const float*)d_in[12];
  const float* fb2 = (const float*)d_in[13];
  float* outp = (float*)d_out;

  float* f1p = (float*)d_ws;                               // 1 MB
  float* f2p = (float*)((char*)d_ws + (size_t)(1 << 20));  // 64 KB

  pack_weights<<<544, 256, 0, stream>>>(f1, f2, f1p, f2p);
  gpc_main<<<dim3(Nn / Kn, Bn), 256, 0, stream>>>(
      points, feats, pmask, w1, b1, w2, b2, w3, b3, fb1, fb2, f1p, f2p, outp);
}